// MeshesRenderer_80960133529998
// MI455X (gfx1250) — compile-verified
//
#include <hip/hip_runtime.h>
#include <math.h>

// ---------------------------------------------------------------------------
// SoftRas forward renderer for gfx1250 (MI455X).
// B=2, M=2, V=256, F=256 (512 faces, 512 verts global), H=W=96, K=16.
//
// Design (see analysis): ~1 GFLOP total -> latency/VALU bound, not HBM bound.
// - Phase 1: vertex transform via V_WMMA_F32_16X16X4_F32 (the one dense
//   matmul in the problem: combined affine 4x3 map applied to 16-vertex tiles).
// - Phase 2: per-face record (projected verts, edges, inv_den, bbox padded by
//   sqrt(BLUR), vertex colors) staged in LDS.
// - Phase 3: per-pixel loop over 512 faces with conservative bbox reject
//   (one ds_load_b128 + 4 cmps kills ~95% of pairs), register top-16 by z.
// - Phase 4: sigmoid/softmax aggregation identical to the reference.
// ---------------------------------------------------------------------------

typedef __attribute__((ext_vector_type(2))) float v2f;
typedef __attribute__((ext_vector_type(8))) float v8f;

#define NV 512
#define NF 512
#define KTOP 16

__device__ __forceinline__ float edge_d2(float px, float py,
                                         float ax, float ay,
                                         float bx, float by_) {
  float abx = bx - ax, aby = by_ - ay;
  float apx = px - ax, apy = py - ay;
  float t = (apx * abx + apy * aby) / (abx * abx + aby * aby + 1e-12f);
  t = fminf(fmaxf(t, 0.0f), 1.0f);
  float dx = apx - t * abx, dy = apy - t * aby;
  return dx * dx + dy * dy;
}

__global__ __launch_bounds__(256) void softras_kernel(
    const float* __restrict__ mpos,    // (B,M,3)
    const float* __restrict__ mrot,    // (B,M,3,3)
    const float* __restrict__ mscale,  // (B,M,3)
    const float* __restrict__ cpos,    // (B,3)
    const float* __restrict__ crot,    // (B,3,3)
    const float* __restrict__ verts,   // (M,V,3) -> (512,3)
    const int*   __restrict__ faces,   // (M,F,3) int32 -> (512,3)
    const float* __restrict__ vcol,    // (M,V,3) -> (512,3)
    float* __restrict__ out)           // (B,96,96,4)
{
  __shared__ float sX[NV], sY[NV], sZ[NV];   // cam-space verts (this batch)
  __shared__ float sRec[NF * 16];            // face records
  __shared__ float sCol[NF * 12];            // face vertex colors (9 used)

  const int tid   = threadIdx.x;
  const int b     = blockIdx.x / 36;      // batch
  const int tile  = blockIdx.x % 36;      // 6x6 tiles of 16x16 pixels
  const int tileX = tile % 6, tileY = tile / 6;

  // ===================== Phase 1: WMMA vertex transform ====================
  // cam_i = sum_j camR[j][i] * ( R[j][k]*s[k]*v_k + pos_j ) + campos_i
  // Combined B (4x16): rows K=0..2 = L[n][k], row K=3 = translation T[n].
  {
    const int  lane = tid & 31;
    const int  wv   = tid >> 5;           // 8 waves
    const int  n    = lane & 15;          // output column N
    const bool lo   = lane < 16;
    const int  nc   = (n < 2) ? n : 2;    // clamped for safe loads
    const int  m    = wv >> 2;            // waves 0-3 mesh 0, 4-7 mesh 1
    const int  rbase = (b * 2 + m) * 9;
    const int  sb    = (b * 2 + m) * 3;

    // camera rotation column nc (einsum 'bvj,bji': cam_i = sum_j w_j*crot[j][i])
    float cr0 = crot[(b * 3 + 0) * 3 + nc];
    float cr1 = crot[(b * 3 + 1) * 3 + nc];
    float cr2 = crot[(b * 3 + 2) * 3 + nc];

    // B-matrix halves pattern (matches documented A / f16-B layouts):
    // VGPR0: lanes0-15 -> K=0, lanes16-31 -> K=2 ; VGPR1: K=1 / K=3
    int   k0 = lo ? 0 : 2;
    float L0 = mscale[sb + k0] * (cr0 * mrot[rbase + 0 + k0] +
                                  cr1 * mrot[rbase + 3 + k0] +
                                  cr2 * mrot[rbase + 6 + k0]);
    float L1 = mscale[sb + 1] * (cr0 * mrot[rbase + 0 + 1] +
                                 cr1 * mrot[rbase + 3 + 1] +
                                 cr2 * mrot[rbase + 6 + 1]);
    float T  = cr0 * mpos[sb + 0] + cr1 * mpos[sb + 1] + cr2 * mpos[sb + 2] +
               cpos[b * 3 + nc];
    float bb0 = (n < 3) ? L0 : 0.0f;
    float bb1 = (n < 3) ? (lo ? L1 : T) : 0.0f;
    v2f Bm = {bb0, bb1};

#pragma unroll
    for (int c = 0; c < 4; ++c) {
      int vb = (wv * 4 + c) * 16;   // 16-vertex chunk base (within mesh m)
      int gv = vb + n;
      // A 16x4: lanes0-15 hold K=0,1 (x,y); lanes16-31 hold K=2,3 (z,1)
      float ax = verts[gv * 3 + 0];
      float ay = verts[gv * 3 + 1];
      float az = verts[gv * 3 + 2];
      v2f Am = {lo ? ax : az, lo ? ay : 1.0f};
      v8f Cm = {};
      v8f Dm = __builtin_amdgcn_wmma_f32_16x16x4_f32(
          false, Am, false, Bm, (short)0, Cm, false, false);
      // D: lane N<16 holds rows M=0..7, lane N+16 rows M=8..15 (column N)
      if (n < 3) {
        float* dst = (n == 0) ? sX : (n == 1) ? sY : sZ;
        int rb = vb + (lo ? 0 : 8);
#pragma unroll
        for (int r = 0; r < 8; ++r) dst[rb + r] = Dm[r];
      }
    }
  }
  __syncthreads();

  // ======================= Phase 2: face setup =============================
  const float PAD = 0.03045f;  // sqrt(BLUR)=0.030349 + margin (conservative)
#pragma unroll
  for (int rep = 0; rep < 2; ++rep) {
    int F = tid + rep * 256;
    int moff = (F >> 8) << 8;  // mesh vertex offset (m*256)
    int i0 = faces[F * 3 + 0] + moff;
    int i1 = faces[F * 3 + 1] + moff;
    int i2 = faces[F * 3 + 2] + moff;
    float z0 = sZ[i0], z1 = sZ[i1], z2 = sZ[i2];
    float p0x = sX[i0] / z0, p0y = sY[i0] / z0;
    float p1x = sX[i1] / z1, p1y = sY[i1] / z1;
    float p2x = sX[i2] / z2, p2y = sY[i2] / z2;
    float e1x = p1x - p0x, e1y = p1y - p0y;
    float e2x = p2x - p0x, e2y = p2y - p0y;
    float invden = 1.0f / (e1x * e2y - e1y * e2x + 1e-12f);
    float bminx = fminf(p0x, fminf(p1x, p2x)) - PAD;
    float bminy = fminf(p0y, fminf(p1y, p2y)) - PAD;
    float bmaxx = fmaxf(p0x, fmaxf(p1x, p2x)) + PAD;
    float bmaxy = fmaxf(p0y, fmaxf(p1y, p2y)) + PAD;
    float* r = &sRec[F * 16];
    r[0] = bminx; r[1] = bminy; r[2]  = bmaxx; r[3]  = bmaxy;
    r[4] = p0x;   r[5] = p0y;   r[6]  = e1x;   r[7]  = e1y;
    r[8] = e2x;   r[9] = e2y;   r[10] = z0;    r[11] = z1;
    r[12] = z2;   r[13] = invden; r[14] = 0.0f; r[15] = 0.0f;
    float* cc = &sCol[F * 12];
#pragma unroll
    for (int c = 0; c < 3; ++c) {
      cc[c]     = vcol[i0 * 3 + c];
      cc[3 + c] = vcol[i1 * 3 + c];
      cc[6 + c] = vcol[i2 * 3 + c];
    }
  }
  __syncthreads();

  // ================= Phase 3: per-pixel loop + top-16 ======================
  // Wave covers an 8x4 pixel tile for coherent bbox accept/reject.
  int lx = (tid & 7) | (((tid >> 5) & 1) << 3);
  int ly = ((tid >> 3) & 3) | (((tid >> 6) & 3) << 2);
  int gx = tileX * 16 + lx, gy = tileY * 16 + ly;
  const float px = (gx + 0.5f) * (2.0f / 96.0f) - 1.0f;
  const float py = 1.0f - (gy + 0.5f) * (2.0f / 96.0f);

  const float BLUR = 9.2102404e-4f;  // log(1/1e-4 - 1) * 1e-4

  float topZ[KTOP], topD[KTOP];
  int   topI[KTOP];
#pragma unroll
  for (int k = 0; k < KTOP; ++k) {
    topZ[k] = __builtin_inff(); topD[k] = 0.0f; topI[k] = -1;
  }

  for (int f = 0; f < NF; ++f) {
    const float4* r4 = (const float4*)&sRec[f * 16];
    float4 bb = r4[0];                       // broadcast ds_load_b128
    if (px < bb.x || py < bb.y || px > bb.z || py > bb.w) continue;

    float4 ra = r4[1];  // v0x v0y e1x e1y
    float4 rb = r4[2];  // e2x e2y z0  z1
    float4 rc = r4[3];  // z2  invden
    float dx = px - ra.x, dy = py - ra.y;
    float b1 = (dx * rb.y - dy * rb.x) * rc.y;
    float b2 = (ra.z * dy - ra.w * dx) * rc.y;
    float b0 = 1.0f - b1 - b2;
    float zpix = b0 * rb.z + b1 * rb.w + b2 * rc.x;
    bool inside = (b0 >= 0.0f) && (b1 >= 0.0f) && (b2 >= 0.0f);
    float v1x = ra.x + ra.z, v1y = ra.y + ra.w;
    float v2x = ra.x + rb.x, v2y = ra.y + rb.y;
    float d2 = fminf(fminf(edge_d2(px, py, ra.x, ra.y, v1x, v1y),
                           edge_d2(px, py, v1x, v1y, v2x, v2y)),
                     edge_d2(px, py, v2x, v2y, ra.x, ra.y));
    float dist = inside ? -d2 : d2;
    bool valid = (zpix > 0.01f) && (dist <= BLUR);
    if (valid && zpix < topZ[KTOP - 1]) {
      topZ[KTOP - 1] = zpix; topD[KTOP - 1] = dist; topI[KTOP - 1] = f;
#pragma unroll
      for (int j = KTOP - 1; j > 0; --j) {   // stable bubble (strict <)
        if (topZ[j] < topZ[j - 1]) {
          float tz = topZ[j]; topZ[j] = topZ[j - 1]; topZ[j - 1] = tz;
          float td = topD[j]; topD[j] = topD[j - 1]; topD[j - 1] = td;
          int   ti = topI[j]; topI[j] = topI[j - 1]; topI[j - 1] = ti;
        }
      }
    }
  }

  // ===================== Phase 4: aggregation ==============================
  const float SIGMA = 1e-4f, GAMMA = 1e-4f, EPS = 1e-10f;
  const float ZINV_SCALE = 1.0f / 99.0f;  // (ZFAR-ZNEAR)=99, ZFAR=100

  float zmax = EPS;
#pragma unroll
  for (int k = 0; k < KTOP; ++k)
    if (topI[k] >= 0) zmax = fmaxf(zmax, (100.0f - topZ[k]) * ZINV_SCALE);

  float accr = 0.0f, accg = 0.0f, accb = 0.0f, wsum = 0.0f, aprod = 1.0f;
#pragma unroll
  for (int k = 0; k < KTOP; ++k) {
    if (topI[k] < 0) continue;   // fillers: mk=0 -> zero contribution
    int f = topI[k];
    float prob = 1.0f / (1.0f + expf(topD[k] * (1.0f / SIGMA)));
    float zinv = (100.0f - topZ[k]) * ZINV_SCALE;
    float w = prob * expf((zinv - zmax) * (1.0f / GAMMA));
    const float4* r4 = (const float4*)&sRec[f * 16];
    float4 ra = r4[1], rb = r4[2], rc = r4[3];
    float dx = px - ra.x, dy = py - ra.y;
    float b1 = (dx * rb.y - dy * rb.x) * rc.y;
    float b2 = (ra.z * dy - ra.w * dx) * rc.y;
    float b0 = 1.0f - b1 - b2;
    const float* cc = &sCol[f * 12];
    accr += w * (b0 * cc[0] + b1 * cc[3] + b2 * cc[6]);
    accg += w * (b0 * cc[1] + b1 * cc[4] + b2 * cc[7]);
    accb += w * (b0 * cc[2] + b1 * cc[5] + b2 * cc[8]);
    wsum += w;
    aprod *= (1.0f - prob);
  }
  float delta = expf((EPS - zmax) * (1.0f / GAMMA));
  float inv = 1.0f / (wsum + delta);
  int o = ((b * 96 + gy) * 96 + gx) * 4;
  out[o + 0] = (accr + delta) * inv;
  out[o + 1] = (accg + delta) * inv;
  out[o + 2] = (accb + delta) * inv;
  out[o + 3] = 1.0f - aprod;
}

extern "C" void kernel_launch(void* const* d_in, const int* in_sizes, int n_in,
                              void* d_out, int out_size, void* d_ws, size_t ws_size,
                              hipStream_t stream) {
  (void)in_sizes; (void)n_in; (void)out_size; (void)d_ws; (void)ws_size;
  const float* mpos   = (const float*)d_in[0];
  const float* mrot   = (const float*)d_in[1];
  const float* mscale = (const float*)d_in[2];
  const float* cpos   = (const float*)d_in[3];
  const float* crot   = (const float*)d_in[4];
  const float* verts  = (const float*)d_in[5];
  const int*   faces  = (const int*)d_in[6];
  const float* vcol   = (const float*)d_in[7];
  float* out = (float*)d_out;
  // 2 batches * 6x6 tiles of 16x16 pixels = 72 blocks, 256 threads (8 waves)
  softras_kernel<<<dim3(72), dim3(256), 0, stream>>>(
      mpos, mrot, mscale, cpos, crot, verts, faces, vcol, out);
}